// QuantumLayer_90752658964948
// MI455X (gfx1250) — compile-verified
//
#include <hip/hip_runtime.h>

typedef float v2f __attribute__((ext_vector_type(2)));
typedef float v8f __attribute__((ext_vector_type(8)));

#define NQ 4
#define NL 2

// ---------------------------------------------------------------------------
// Kernel 1: build the 16x16 circuit unitary U from q_weights.
// Thread j (j<16) evolves basis state e_j through 2 layers of Rot+CNOT;
// final state is column j of U. Ur -> ws[0..255], Ui -> ws[256..511].
// Wire w corresponds to bit (3-w) of the flattened index k.
// ---------------------------------------------------------------------------
__global__ void build_unitary_kernel(const float* __restrict__ qw,
                                     float* __restrict__ Uws) {
    int j = threadIdx.x;
    if (j >= 16) return;
    float sr[16], si[16];
#pragma unroll
    for (int k = 0; k < 16; ++k) { sr[k] = (k == j) ? 1.0f : 0.0f; si[k] = 0.0f; }

#pragma unroll
    for (int l = 0; l < NL; ++l) {
        // Rot(phi,theta,omega) = RZ(omega) RY(theta) RZ(phi) on each wire
#pragma unroll
        for (int w = 0; w < NQ; ++w) {
            float phi = qw[(l * NQ + w) * 3 + 0];
            float th  = qw[(l * NQ + w) * 3 + 1];
            float om  = qw[(l * NQ + w) * 3 + 2];
            float ct = __cosf(0.5f * th), st = __sinf(0.5f * th);
            float a  = 0.5f * (phi + om), bb = 0.5f * (phi - om);
            float ca = __cosf(a),  sa = __sinf(a);
            float cb = __cosf(bb), sb = __sinf(bb);
            // U = [[e^{-ia}ct, -e^{ib}st],[e^{-ib}st, e^{ia}ct]]
            float u00r =  ca * ct, u00i = -sa * ct;
            float u01r = -cb * st, u01i = -sb * st;
            float u10r =  cb * st, u10i = -sb * st;
            float u11r =  ca * ct, u11i =  sa * ct;
            int bit = 1 << (3 - w);
#pragma unroll
            for (int k0 = 0; k0 < 16; ++k0) {
                if (k0 & bit) continue;
                int k1 = k0 | bit;
                float a0r = sr[k0], a0i = si[k0];
                float a1r = sr[k1], a1i = si[k1];
                sr[k0] = u00r * a0r - u00i * a0i + u01r * a1r - u01i * a1i;
                si[k0] = u00r * a0i + u00i * a0r + u01r * a1i + u01i * a1r;
                sr[k1] = u10r * a0r - u10i * a0i + u11r * a1r - u11i * a1i;
                si[k1] = u10r * a0i + u10i * a0r + u11r * a1i + u11i * a1r;
            }
        }
        int r = l % (NQ - 1) + 1;  // StronglyEntanglingLayers default range
#pragma unroll
        for (int w = 0; w < NQ; ++w) {
            int tw = (w + r) & 3;
            int cbit = 1 << (3 - w);
            int tbit = 1 << (3 - tw);
#pragma unroll
            for (int k = 0; k < 16; ++k) {
                if ((k & cbit) && !(k & tbit)) {
                    int k2 = k | tbit;
                    float tr = sr[k]; sr[k] = sr[k2]; sr[k2] = tr;
                    float ti = si[k]; si[k] = si[k2]; si[k2] = ti;
                }
            }
        }
    }
#pragma unroll
    for (int k = 0; k < 16; ++k) {
        Uws[k * 16 + j]       = sr[k];
        Uws[256 + k * 16 + j] = si[k];
    }
}

// ---------------------------------------------------------------------------
// Kernel 2: fused pre-GEMM -> quantum matvec (WMMA) -> post-GEMM (WMMA).
// 256 threads = 8 waves; each wave owns 32 consecutive batch rows (same n,
// consecutive w in HW plane -> coalesced x loads and vectorized out stores).
// ---------------------------------------------------------------------------
__global__ __launch_bounds__(256) void qlayer_fused_kernel(
    const float* __restrict__ x,      // (8,128,128,128)
    const float* __restrict__ pre_w,  // (4,128)
    const float* __restrict__ pre_b,  // (4)
    const float* __restrict__ Uws,    // Ur[256], Ui[256]
    const float* __restrict__ post_w, // (128,4)
    const float* __restrict__ post_b, // (128)
    float* __restrict__ out)          // (8,128,128,128)
{
    __shared__ float prew[512];        // pre_w transposed: [c][q]
    __shared__ float psiT[8 * 512];    // per-wave psi0 transpose: [j(16)][row(32)]

    const int tid   = threadIdx.x;
    const int lane  = tid & 31;
    const int wid   = tid >> 5;
    const int b     = blockIdx.x * 256 + tid;   // this thread's batch row
    const int row16 = lane & 15;
    const int sel   = (lane < 16) ? 0 : 2;      // K-half selector (WMMA A/B layout)

    // Stage pre_w transposed into LDS (2 elements per thread)
    {
        int i0 = tid, i1 = tid + 256;
        prew[i0] = pre_w[(i0 & 3) * 128 + (i0 >> 2)];
        prew[i1] = pre_w[(i1 & 3) * 128 + (i1 >> 2)];
    }
    __syncthreads();

    // ---- Phase 1: pre-GEMM (coalesced stream over x, 4 FMAs per load) ----
    const int n  = b >> 14;      // H*W = 16384
    const int hw = b & 16383;
    const float* xb = x + (size_t)n * 2097152 + hw;

    __builtin_prefetch(xb,               0, 0);
    __builtin_prefetch(xb + (32 << 14),  0, 0);
    __builtin_prefetch(xb + (64 << 14),  0, 0);
    __builtin_prefetch(xb + (96 << 14),  0, 0);

    float acc0 = pre_b[0], acc1 = pre_b[1], acc2 = pre_b[2], acc3 = pre_b[3];
    const float4* pw4 = (const float4*)prew;
#pragma unroll 4
    for (int c = 0; c < 128; ++c) {
        float  xv = xb[c << 14];
        float4 w4 = pw4[c];
        acc0 = fmaf(xv, w4.x, acc0);
        acc1 = fmaf(xv, w4.y, acc1);
        acc2 = fmaf(xv, w4.z, acc2);
        acc3 = fmaf(xv, w4.w, acc3);
    }

    // ---- Phase 2: product state psi0, transpose into WMMA B layout -------
    const float HPI = 1.5707963267948966f;
    float c0 = __cosf(HPI * acc0), s0 = __sinf(HPI * acc0);
    float c1 = __cosf(HPI * acc1), s1 = __sinf(HPI * acc1);
    float c2 = __cosf(HPI * acc2), s2 = __sinf(HPI * acc2);
    float c3 = __cosf(HPI * acc3), s3 = __sinf(HPI * acc3);
    float pa[4] = { c0 * c1, c0 * s1, s0 * c1, s0 * s1 };  // bits b0b1 (k>>2)
    float pb[4] = { c2 * c3, c2 * s3, s2 * c3, s2 * s3 };  // bits b2b3 (k&3)

    float* myT = psiT + wid * 512;
#pragma unroll
    for (int j = 0; j < 16; ++j)
        myT[j * 32 + lane] = pa[j >> 2] * pb[j & 3];
    __syncthreads();

    // ---- Phase 3: amp = U * psi0 via chained V_WMMA_F32_16X16X4_F32 ------
    // A layout (16x4): lanes 0-15 hold M=lane K={0,1}; lanes 16-31 K={2,3}.
    v2f aR[4], aI[4];
#pragma unroll
    for (int kk = 0; kk < 4; ++kk) {
        aR[kk] = *(const v2f*)(Uws +       row16 * 16 + 4 * kk + sel);
        aI[kk] = *(const v2f*)(Uws + 256 + row16 * 16 + 4 * kk + sel);
    }

    v2f zA[2];  // post-GEMM A operand (z in A layout), one per 16-row half
#pragma unroll
    for (int h = 0; h < 2; ++h) {
        v8f accR = {0.f,0.f,0.f,0.f,0.f,0.f,0.f,0.f};
        v8f accI = {0.f,0.f,0.f,0.f,0.f,0.f,0.f,0.f};
        const float* base = myT + h * 16 + row16;  // B: lane -> column (batch row)
#pragma unroll
        for (int kk = 0; kk < 4; ++kk) {
            int j0 = 4 * kk + sel;
            v2f bv;
            bv.x = base[j0 * 32];
            bv.y = base[(j0 + 1) * 32];
            accR = __builtin_amdgcn_wmma_f32_16x16x4_f32(
                false, aR[kk], false, bv, (short)0, accR, false, false);
            accI = __builtin_amdgcn_wmma_f32_16x16x4_f32(
                false, aI[kk], false, bv, (short)0, accI, false, false);
        }
        // probabilities + signed PauliZ reductions (wire w <-> bit 3-w)
        float zp0 = 0.f, zp1 = 0.f, zp2 = 0.f, zp3 = 0.f;
        int kbase = (lane < 16) ? 0 : 8;   // C layout: vgpr r -> M = kbase+r
#pragma unroll
        for (int r2 = 0; r2 < 8; ++r2) {
            float p = accR[r2] * accR[r2] + accI[r2] * accI[r2];
            int k = kbase + r2;
            zp0 += ((k >> 3) & 1) ? -p : p;
            zp1 += ((k >> 2) & 1) ? -p : p;
            zp2 += ((k >> 1) & 1) ? -p : p;
            zp3 += ( k       & 1) ? -p : p;
        }
        // combine the two k-halves held by lane pair (l, l^16)
        zp0 += __shfl_xor(zp0, 16, 32);
        zp1 += __shfl_xor(zp1, 16, 32);
        zp2 += __shfl_xor(zp2, 16, 32);
        zp3 += __shfl_xor(zp3, 16, 32);
        // z is now exactly in A layout: lanes<16 carry K={0,1}, lanes>=16 K={2,3}
        v2f za;
        za.x = (sel == 0) ? zp0 : zp2;
        za.y = (sel == 0) ? zp1 : zp3;
        zA[h] = za;
    }

    // ---- Phase 4: post-GEMM out = z @ post_w^T + post_b via WMMA ---------
    const int b0row = blockIdx.x * 256 + wid * 32;  // wave's first batch row
    const int n2  = b0row >> 14;
    const int hw2 = b0row & 16383;
    float* outw = out + (size_t)n2 * 2097152 + hw2;
    const float4* pwt = (const float4*)post_w;

#pragma unroll
    for (int t = 0; t < 8; ++t) {
        // B layout (4x16): lane -> N = t*16+row16; VGPR0/1 -> K = sel, sel+1
        float4 pw = pwt[t * 16 + row16];
        v2f bt;
        bt.x = (sel == 0) ? pw.x : pw.z;
        bt.y = (sel == 0) ? pw.y : pw.w;
        float bias = post_b[t * 16 + row16];
        v8f C = {bias, bias, bias, bias, bias, bias, bias, bias};

        v8f D0 = __builtin_amdgcn_wmma_f32_16x16x4_f32(
            false, zA[0], false, bt, (short)0, C, false, false);  // rows 0-15
        v8f D1 = __builtin_amdgcn_wmma_f32_16x16x4_f32(
            false, zA[1], false, bt, (short)0, C, false, false);  // rows 16-31

        // D layout: lane holds 8 consecutive-M values of column N = t*16+row16,
        // M maps to consecutive w -> contiguous float4 stores.
        float* op = outw + (size_t)(t * 16 + row16) * 16384 + ((lane < 16) ? 0 : 8);
        *(float4*)(op)      = make_float4(D0[0], D0[1], D0[2], D0[3]);
        *(float4*)(op + 4)  = make_float4(D0[4], D0[5], D0[6], D0[7]);
        *(float4*)(op + 16) = make_float4(D1[0], D1[1], D1[2], D1[3]);
        *(float4*)(op + 20) = make_float4(D1[4], D1[5], D1[6], D1[7]);
    }
}

extern "C" void kernel_launch(void* const* d_in, const int* in_sizes, int n_in,
                              void* d_out, int out_size, void* d_ws, size_t ws_size,
                              hipStream_t stream) {
    const float* x      = (const float*)d_in[0];
    const float* pre_w  = (const float*)d_in[1];
    const float* pre_b  = (const float*)d_in[2];
    const float* qw     = (const float*)d_in[3];
    const float* post_w = (const float*)d_in[4];
    const float* post_b = (const float*)d_in[5];
    float* out = (float*)d_out;
    float* Uws = (float*)d_ws;   // needs 512 floats

    build_unitary_kernel<<<1, 32, 0, stream>>>(qw, Uws);

    const int B = in_sizes[0] / 128;          // N*H*W = 131072 batch rows
    qlayer_fused_kernel<<<B / 256, 256, 0, stream>>>(
        x, pre_w, pre_b, Uws, post_w, post_b, out);
}